// HungarianMatcher_14568529068591
// MI455X (gfx1250) — compile-verified
//
#include <hip/hip_runtime.h>

typedef __attribute__((ext_vector_type(2))) float v2f;
typedef __attribute__((ext_vector_type(8))) float v8f;

#define K_COST_SPAN 5.0f
#define K_COST_GIOU 2.0f
#define K_EPS 1e-6f
#define NC 16

// ---------------------------------------------------------------------------
// Prep kernel 1: negated softmax per query row -> d_ws  (A operand source)
// One thread per query: 64B in, 64B out.  8192 threads total.
// ---------------------------------------------------------------------------
__global__ __launch_bounds__(256) void prep_negprob_kernel(
    const float* __restrict__ logits,   // [BQ][16]
    float*       __restrict__ negprob)  // [BQ][16]
{
    const unsigned q = blockIdx.x * blockDim.x + threadIdx.x;
    const float4* lp = (const float4*)(logits + (q << 4));
    float4 L0 = lp[0], L1 = lp[1], L2 = lp[2], L3 = lp[3];
    float l[NC] = { L0.x, L0.y, L0.z, L0.w,  L1.x, L1.y, L1.z, L1.w,
                    L2.x, L2.y, L2.z, L2.w,  L3.x, L3.y, L3.z, L3.w };
    float mx = l[0];
#pragma unroll
    for (int c = 1; c < NC; ++c) mx = fmaxf(mx, l[c]);
    float p[NC];
    float sum = 0.0f;
#pragma unroll
    for (int c = 0; c < NC; ++c) { p[c] = __expf(l[c] - mx); sum += p[c]; }
    const float ninv = -1.0f / sum;     // A = -prob  (COST_CLASS == 1)
#pragma unroll
    for (int c = 0; c < NC; ++c) p[c] *= ninv;
    float4* op = (float4*)(negprob + (q << 4));
    op[0] = make_float4(p[0],  p[1],  p[2],  p[3]);
    op[1] = make_float4(p[4],  p[5],  p[6],  p[7]);
    op[2] = make_float4(p[8],  p[9],  p[10], p[11]);
    op[3] = make_float4(p[12], p[13], p[14], p[15]);
}

// ---------------------------------------------------------------------------
// Prep kernel 2: one-hot rows of tgt_ids -> d_ws  (B operand source)
// One thread per target: 64B out.  2048 threads total.
// ---------------------------------------------------------------------------
__global__ __launch_bounds__(256) void prep_onehot_kernel(
    const int* __restrict__ tids,       // [NT]
    float*     __restrict__ onehot)     // [NT][16]
{
    const unsigned t  = blockIdx.x * blockDim.x + threadIdx.x;
    const int id = tids[t];
    float4* op = (float4*)(onehot + (t << 4));
#pragma unroll
    for (int c4 = 0; c4 < 4; ++c4) {
        op[c4] = make_float4((id == 4 * c4 + 0) ? 1.0f : 0.0f,
                             (id == 4 * c4 + 1) ? 1.0f : 0.0f,
                             (id == 4 * c4 + 2) ? 1.0f : 0.0f,
                             (id == 4 * c4 + 3) ? 1.0f : 0.0f);
    }
}

// ---------------------------------------------------------------------------
// Main kernel: one wave32 computes one 16x16 tile of the cost matrix.
//   A (16x4 f32 / WMMA)  loaded b64 from negprob in native A layout
//   B (4x16 f32 / WMMA)  loaded b64 from onehot  in native B layout
//   C (16x16 f32)        = 5*cost_span - 2*giou computed in VALU in C/D layout
//   D = A*B + C          via 4 chained v_wmma_f32_16x16x4_f32
// All buffers < 4GB: each stream gets ONE 64-bit pointer add from a 32-bit
// element offset, then every load/store uses a 24-bit immediate offset.
// ---------------------------------------------------------------------------
template<int NT>
__global__ __launch_bounds__(256) void matcher_cost_main_kernel(
    const float* __restrict__ negprob,  // [BQ][16]
    const float* __restrict__ onehot,   // [NT][16]
    const float* __restrict__ pspans,   // [BQ][2]
    const float* __restrict__ tspans,   // [NT][2]
    float*       __restrict__ out)      // [BQ][NT]
{
    constexpr unsigned NTT = (unsigned)NT >> 4;       // t-tiles per q-row
    const unsigned lane = threadIdx.x & 31u;
    const unsigned wave = threadIdx.x >> 5;
    const unsigned tile = blockIdx.x * 8u + wave;     // 8 waves per block
    const unsigned tq   = tile / NTT;                 // power-of-two -> shift
    const unsigned tt   = tile % NTT;
    const unsigned q0   = tq << 4;
    const unsigned t0   = tt << 4;
    const unsigned hi   = lane >> 4;                  // 0: lanes 0-15, 1: 16-31
    const unsigned ln   = lane & 15u;
    const unsigned t    = t0 + ln;

    // A operand: lane needs negprob[q0+ln][4*kk + 2*hi .. +1]  (8B aligned)
    const float* ap = negprob + (((q0 + ln) << 4) + (hi << 1));  // 1 ptr add
    v2f a0 = *(const v2f*)(ap + 0);                              // imm 0
    v2f a1 = *(const v2f*)(ap + 4);                              // imm 16
    v2f a2 = *(const v2f*)(ap + 8);                              // imm 32
    v2f a3 = *(const v2f*)(ap + 12);                             // imm 48

    // B operand: lane needs onehot[t][4*kk + 2*hi .. +1]
    const float* bp = onehot + ((t << 4) + (hi << 1));           // 1 ptr add
    v2f b0 = *(const v2f*)(bp + 0);
    v2f b1 = *(const v2f*)(bp + 4);
    v2f b2 = *(const v2f*)(bp + 8);
    v2f b3 = *(const v2f*)(bp + 12);

    // C accumulator: span + giou costs in WMMA C/D layout
    const float2 ts = *(const float2*)(tspans + (t << 1));
    const float s2 = ts.x, e2 = ts.y, d2 = e2 - s2;

    const float2* qsp = (const float2*)pspans + (q0 + (hi << 3)); // 1 ptr add
    v8f acc;
#pragma unroll
    for (int j = 0; j < 8; ++j) {                     // VGPR j <-> M = j + 8*hi
        const float2 qs = qsp[j];                     // imm offsets j*8
        const float s1 = qs.x, e1 = qs.y, d1 = e1 - s1;
        const float cost_span = fabsf(s1 - s2) + fabsf(e1 - e2);
        const float inter = fmaxf(0.0f, fminf(e1, e2) - fmaxf(s1, s2));
        const float uni   = d1 + d2 - inter;
        const float iou   = inter / (uni + K_EPS);
        const float enc   = fmaxf(e1, e2) - fminf(s1, s2);
        const float giou  = iou - (enc - uni) / (enc + K_EPS);
        acc[j] = K_COST_SPAN * cost_span - K_COST_GIOU * giou;
    }

    // D = (-P) x OneHot + C : 4 chained v_wmma_f32_16x16x4_f32
    acc = __builtin_amdgcn_wmma_f32_16x16x4_f32(false, a0, false, b0, (short)0, acc, false, false);
    acc = __builtin_amdgcn_wmma_f32_16x16x4_f32(false, a1, false, b1, (short)0, acc, false, false);
    acc = __builtin_amdgcn_wmma_f32_16x16x4_f32(false, a2, false, b2, (short)0, acc, false, false);
    acc = __builtin_amdgcn_wmma_f32_16x16x4_f32(false, a3, false, b3, (short)0, acc, false, false);

    // Store tile: one ptr add, then 8 stores with immediate row offsets
    // (j*NT*4 bytes, max 57344 < 2^23).
    float* o = out + ((q0 + (hi << 3)) * (unsigned)NT + t);      // 1 ptr add
#pragma unroll
    for (int j = 0; j < 8; ++j)
        o[j * NT] = acc[j];
}

// ---------------------------------------------------------------------------
// Fallback: fully inline kernel (generic nt, no workspace needed).
// ---------------------------------------------------------------------------
__global__ __launch_bounds__(256) void matcher_cost_inline_kernel(
    const float* __restrict__ logits, const float* __restrict__ pspans,
    const float* __restrict__ tspans, const int* __restrict__ tids,
    float* __restrict__ out, int nt)
{
    const int lane = threadIdx.x & 31;
    const int wave = threadIdx.x >> 5;
    const int ntt  = nt >> 4;
    const int tile = blockIdx.x * 8 + wave;
    const int tq = tile / ntt, tt = tile - tq * ntt;
    const int q0 = tq << 4, t0 = tt << 4;
    const int hi = lane >> 4, ln = lane & 15;

    const float4* lp = (const float4*)(logits + (size_t)(q0 + ln) * NC);
    float4 L0 = lp[0], L1 = lp[1], L2 = lp[2], L3 = lp[3];
    float l[NC] = { L0.x, L0.y, L0.z, L0.w,  L1.x, L1.y, L1.z, L1.w,
                    L2.x, L2.y, L2.z, L2.w,  L3.x, L3.y, L3.z, L3.w };
    float mx = l[0];
#pragma unroll
    for (int c = 1; c < NC; ++c) mx = fmaxf(mx, l[c]);
    float p[NC]; float sum = 0.0f;
#pragma unroll
    for (int c = 0; c < NC; ++c) { p[c] = __expf(l[c] - mx); sum += p[c]; }
    const float ninv = -1.0f / sum;
#pragma unroll
    for (int c = 0; c < NC; ++c) p[c] *= ninv;

    const bool h = (hi != 0);
    v2f a0 = { h ? p[ 2] : p[ 0],  h ? p[ 3] : p[ 1] };
    v2f a1 = { h ? p[ 6] : p[ 4],  h ? p[ 7] : p[ 5] };
    v2f a2 = { h ? p[10] : p[ 8],  h ? p[11] : p[ 9] };
    v2f a3 = { h ? p[14] : p[12],  h ? p[15] : p[13] };

    const int t  = t0 + ln;
    const int id = tids[t];
    const int kb = hi << 1;
    v2f b0 = { (id == kb +  0) ? 1.0f : 0.0f, (id == kb +  1) ? 1.0f : 0.0f };
    v2f b1 = { (id == kb +  4) ? 1.0f : 0.0f, (id == kb +  5) ? 1.0f : 0.0f };
    v2f b2 = { (id == kb +  8) ? 1.0f : 0.0f, (id == kb +  9) ? 1.0f : 0.0f };
    v2f b3 = { (id == kb + 12) ? 1.0f : 0.0f, (id == kb + 13) ? 1.0f : 0.0f };

    const float2 ts = ((const float2*)tspans)[t];
    const float s2 = ts.x, e2 = ts.y, d2 = e2 - s2;
    v8f acc;
#pragma unroll
    for (int j = 0; j < 8; ++j) {
        const int q = q0 + j + (hi << 3);
        const float2 qs = ((const float2*)pspans)[q];
        const float s1 = qs.x, e1 = qs.y, d1 = e1 - s1;
        const float cost_span = fabsf(s1 - s2) + fabsf(e1 - e2);
        const float inter = fmaxf(0.0f, fminf(e1, e2) - fmaxf(s1, s2));
        const float uni   = d1 + d2 - inter;
        const float iou   = inter / (uni + K_EPS);
        const float enc   = fmaxf(e1, e2) - fminf(s1, s2);
        const float giou  = iou - (enc - uni) / (enc + K_EPS);
        acc[j] = K_COST_SPAN * cost_span - K_COST_GIOU * giou;
    }
    acc = __builtin_amdgcn_wmma_f32_16x16x4_f32(false, a0, false, b0, (short)0, acc, false, false);
    acc = __builtin_amdgcn_wmma_f32_16x16x4_f32(false, a1, false, b1, (short)0, acc, false, false);
    acc = __builtin_amdgcn_wmma_f32_16x16x4_f32(false, a2, false, b2, (short)0, acc, false, false);
    acc = __builtin_amdgcn_wmma_f32_16x16x4_f32(false, a3, false, b3, (short)0, acc, false, false);

    float* orow = out + (size_t)(q0 + (hi << 3)) * nt + t;
#pragma unroll
    for (int j = 0; j < 8; ++j) orow[(size_t)j * nt] = acc[j];
}

extern "C" void kernel_launch(void* const* d_in, const int* in_sizes, int n_in,
                              void* d_out, int out_size, void* d_ws, size_t ws_size,
                              hipStream_t stream) {
    (void)n_in; (void)out_size;
    const float* logits = (const float*)d_in[0];   // pred_logits [bs*nq*16]
    const float* pspans = (const float*)d_in[1];   // pred_spans  [bs*nq*2]
    const float* tspans = (const float*)d_in[2];   // tgt_spans   [nt*2]
    const int*   tids   = (const int*)  d_in[3];   // tgt_ids     [nt]
    float* out = (float*)d_out;

    const int bq = in_sizes[0] / NC;               // bs*nq = 8192
    const int nt = in_sizes[3];                    // 2048
    const int tiles  = (bq >> 4) * (nt >> 4);
    const int blocks = tiles / 8;                  // 8 waves / 256-thread block

    const size_t need = (size_t)bq * NC * sizeof(float)   // negprob
                      + (size_t)nt * NC * sizeof(float);  // onehot

    if (nt == 2048 && (bq % 256) == 0 && (nt % 256) == 0 && ws_size >= need) {
        float* negprob = (float*)d_ws;
        float* onehot  = negprob + (size_t)bq * NC;
        prep_negprob_kernel<<<bq / 256, 256, 0, stream>>>(logits, negprob);
        prep_onehot_kernel<<<nt / 256, 256, 0, stream>>>(tids, onehot);
        matcher_cost_main_kernel<2048><<<blocks, 256, 0, stream>>>(
            negprob, onehot, pspans, tspans, out);
    } else {
        matcher_cost_inline_kernel<<<blocks, 256, 0, stream>>>(
            logits, pspans, tspans, tids, out, nt);
    }
}